// VectorQuantizerEMA_5179730559566
// MI455X (gfx1250) — compile-verified
//
#include <hip/hip_runtime.h>
#include <hip/hip_bf16.h>

typedef float v2f __attribute__((ext_vector_type(2)));
typedef float v8f __attribute__((ext_vector_type(8)));

#define BN      16384      // 8*2048 rows
#define DIM     512
#define KCODES  8192
#define M_PER_WG 128       // 8 waves * 16 rows
#define LDS_STRIDE 516     // 512 + 4 pad -> conflict-free strided float2 reads
#define DECAY_F 0.99f
#define ONE_MINUS_DECAY 0.01f
#define EPS_F 1e-5f

// ---------------------------------------------------------------- utilities

__global__ void zero_f4_kernel(float4* __restrict__ p, long long n4) {
    long long i = (long long)blockIdx.x * blockDim.x + threadIdx.x;
    long long stride = (long long)gridDim.x * blockDim.x;
    float4 z = make_float4(0.f, 0.f, 0.f, 0.f);
    for (; i < n4; i += stride) p[i] = z;
}

__global__ void scale_kernel(const float* __restrict__ in, float* __restrict__ out, long long n) {
    long long i = (long long)blockIdx.x * blockDim.x + threadIdx.x;
    long long stride = (long long)gridDim.x * blockDim.x;
    for (; i < n; i += stride) out[i] = in[i] * DECAY_F;
}

// ---------------------------------------------------------------- ||e_k||^2 (one wave per code row); also zeroes the loss accumulator
__global__ void enorm_kernel(const float* __restrict__ emb, float* __restrict__ enorm,
                             float* __restrict__ sq_sum) {
    if (blockIdx.x == 0 && threadIdx.x == 0) *sq_sum = 0.f;
    int gid  = blockIdx.x * blockDim.x + threadIdx.x;
    int row  = gid >> 5;
    int lane = threadIdx.x & 31;
    if (row >= KCODES) return;
    const float4* p = (const float4*)(emb + (size_t)row * DIM);
    float s = 0.f;
#pragma unroll
    for (int j = 0; j < 4; ++j) {
        float4 v = p[lane + 32 * j];
        s += v.x * v.x + v.y * v.y + v.z * v.z + v.w * v.w;
    }
#pragma unroll
    for (int m = 16; m >= 1; m >>= 1) s += __shfl_xor(s, m, 32);
    if (lane == 0) enorm[row] = s;
}

// ---------------------------------------------------------------- fused distance GEMM (f32 WMMA) + row argmin
// WG = 256 threads = 8 waves. WG stages 128 rows of A in LDS (258 KB); wave w
// owns rows w*16..w*16+15 and sweeps N-tile PAIRS {2w+16j, 2w+16j+1}, j=0..31.
// Two concurrent accumulator chains per wave share the LDS A operands:
// 4 independent WMMA chains per SIMD at 2 waves/SIMD -> hides matrix-pipe latency.
__global__ void vq_argmin_kernel(const float* __restrict__ flat,
                                 const float* __restrict__ emb,
                                 const float* __restrict__ enorm,
                                 int* __restrict__ idx_out) {
    extern __shared__ float lds[];
    const int tid  = threadIdx.x;
    const int wave = tid >> 5;
    const int lane = tid & 31;
    const int hi   = lane >> 4;   // half-wave select (K pair / M+8 per ISA layout)
    const int ln   = lane & 15;
    const int row0 = blockIdx.x * M_PER_WG;

    // Cooperative A load: 128 rows x 512 floats = 16384 float4, coalesced.
    for (int i = tid; i < M_PER_WG * (DIM / 4); i += 256) {
        int r = i >> 7;            // /(512/4)
        int c4 = i & 127;
        float4 v = ((const float4*)(flat + (size_t)(row0 + r) * DIM))[c4];
        *(float4*)(&lds[r * LDS_STRIDE + c4 * 4]) = v;   // row stride 2064B: 16B aligned
    }
    __syncthreads();

    // A operand: lanes0-15 rows M=ln K={k,k+1}; lanes16-31 same rows K={k+2,k+3}
    const float* a_base = &lds[(wave * 16 + ln) * LDS_STRIDE + 2 * hi];

    float best_d0[8], best_d1[8];
    int   best_i0[8], best_i1[8];
#pragma unroll
    for (int i = 0; i < 8; ++i) {
        best_d0[i] = __builtin_inff(); best_i0[i] = 0;
        best_d1[i] = __builtin_inff(); best_i1[i] = 0;
    }

    for (int j = 0; j < 32; ++j) {
        const int n0 = (2 * wave + 16 * j) * 16;         // first tile's code base
        // B operand (4x16 = E^T tile): lane column N=ln, K pair selected by hi
        const float* b_base0 = emb + (size_t)(n0 + ln) * DIM + 2 * hi;
        const float* b_base1 = b_base0 + (size_t)16 * DIM;
        if (j + 1 < 32)
            __builtin_prefetch(emb + (size_t)(n0 + 256 + ln) * DIM, 0, 0);

        v8f c0 = {};
        v8f c1 = {};
#pragma unroll 8
        for (int k = 0; k < DIM; k += 4) {
            v2f a  = *(const v2f*)(a_base + k);
            v2f b0 = *(const v2f*)(b_base0 + k);
            v2f b1 = *(const v2f*)(b_base1 + k);
            c0 = __builtin_amdgcn_wmma_f32_16x16x4_f32(false, a, false, b0,
                                                       (short)0, c0, false, false);
            c1 = __builtin_amdgcn_wmma_f32_16x16x4_f32(false, a, false, b1,
                                                       (short)0, c1, false, false);
        }
        float en0 = enorm[n0 + ln];
        float en1 = enorm[n0 + 16 + ln];
        int nidx0 = n0 + ln;
        int nidx1 = n0 + 16 + ln;
#pragma unroll
        for (int i = 0; i < 8; ++i) {
            float d0 = en0 - 2.0f * c0[i];        // ||x||^2 constant per row: dropped
            float d1 = en1 - 2.0f * c1[i];
            if (d0 < best_d0[i]) { best_d0[i] = d0; best_i0[i] = nidx0; }
            if (d1 < best_d1[i]) { best_d1[i] = d1; best_i1[i] = nidx1; }
        }
    }

    // Merge the two chains (tie-break: lowest code index, like jnp.argmin).
#pragma unroll
    for (int i = 0; i < 8; ++i) {
        if (best_d1[i] < best_d0[i] ||
            (best_d1[i] == best_d0[i] && best_i1[i] < best_i0[i])) {
            best_d0[i] = best_d1[i]; best_i0[i] = best_i1[i];
        }
    }

    // Butterfly min+argmin within each 16-lane half (xor masks 1..8 stay in-half).
#pragma unroll
    for (int m = 1; m <= 8; m <<= 1) {
#pragma unroll
        for (int i = 0; i < 8; ++i) {
            float od = __shfl_xor(best_d0[i], m, 32);
            int   oi = __shfl_xor(best_i0[i], m, 32);
            if (od < best_d0[i] || (od == best_d0[i] && oi < best_i0[i])) {
                best_d0[i] = od; best_i0[i] = oi;
            }
        }
    }
    // C layout: VGPR i -> M=i (lanes0-15) / M=i+8 (lanes16-31)
    if (ln == 0) {
        int mbase = row0 + wave * 16 + hi * 8;
#pragma unroll
        for (int i = 0; i < 8; ++i) idx_out[mbase + i] = best_i0[i];
    }
}

// ---------------------------------------------------------------- gather + loss + scatter (one WG per row)
__global__ void quantize_scatter_kernel(const float* __restrict__ flat,
                                        const float* __restrict__ emb,
                                        const int* __restrict__ idx,
                                        float* __restrict__ q_out,
                                        float* __restrict__ enc_out,
                                        float* __restrict__ ncs,
                                        float* __restrict__ new_ema_w,
                                        float* __restrict__ sq_sum) {
    int row = blockIdx.x;
    int k   = idx[row];
    const float* e = emb  + (size_t)k   * DIM;
    const float* x = flat + (size_t)row * DIM;
    float local = 0.f;
    for (int d = threadIdx.x; d < DIM; d += 256) {
        float ev = e[d], xv = x[d];
        q_out[(size_t)row * DIM + d] = ev;          // straight-through == quantized
        float diff = ev - xv;
        local += diff * diff;
        atomicAdd(&new_ema_w[(size_t)k * DIM + d], ONE_MINUS_DECAY * xv);  // dw EMA term
    }
#pragma unroll
    for (int m = 16; m >= 1; m >>= 1) local += __shfl_xor(local, m, 32);
    __shared__ float warp_s[8];
    if ((threadIdx.x & 31) == 0) warp_s[threadIdx.x >> 5] = local;
    __syncthreads();
    if (threadIdx.x == 0) {
        float s = 0.f;
#pragma unroll
        for (int w = 0; w < 8; ++w) s += warp_s[w];
        atomicAdd(sq_sum, s);
        enc_out[(size_t)row * KCODES + k] = 1.0f;   // one-hot
        atomicAdd(&ncs[k], ONE_MINUS_DECAY);        // (1-decay)*counts EMA term
    }
}

// ---------------------------------------------------------------- cluster-size normalize + loss (single WG)
__global__ void finalize_kernel(float* __restrict__ ncs,
                                const float* __restrict__ sq_sum,
                                float* __restrict__ loss_out) {
    float s = 0.f;
    for (int i = threadIdx.x; i < KCODES; i += 1024) s += ncs[i];
#pragma unroll
    for (int m = 16; m >= 1; m >>= 1) s += __shfl_xor(s, m, 32);
    __shared__ float warp_s[32];
    if ((threadIdx.x & 31) == 0) warp_s[threadIdx.x >> 5] = s;
    __syncthreads();
    __shared__ float n_tot;
    if (threadIdx.x == 0) {
        float t = 0.f;
#pragma unroll
        for (int w = 0; w < 32; ++w) t += warp_s[w];
        n_tot = t;
        loss_out[0] = 0.25f * sq_sum[0] / (float)(BN * DIM);  // commitment * mean
    }
    __syncthreads();
    float n = n_tot;
    float denom = n + (float)KCODES * EPS_F;
    for (int i = threadIdx.x; i < KCODES; i += 1024)
        ncs[i] = (ncs[i] + EPS_F) / denom * n;
}

__global__ void new_weight_kernel(const float* __restrict__ emaw,
                                  const float* __restrict__ ncs,
                                  float* __restrict__ w) {
    long long i = (long long)blockIdx.x * blockDim.x + threadIdx.x;
    if (i < (long long)KCODES * DIM) w[i] = emaw[i] / ncs[i >> 9];  // /512
}

// ---------------------------------------------------------------- launcher

extern "C" void kernel_launch(void* const* d_in, const int* in_sizes, int n_in,
                              void* d_out, int out_size, void* d_ws, size_t ws_size,
                              hipStream_t stream) {
    const float* flat = (const float*)d_in[0];   // [8,2048,512]
    const float* emb  = (const float*)d_in[1];   // [8192,512]
    const float* emaw = (const float*)d_in[2];   // [8192,512]
    const float* ecs  = (const float*)d_in[3];   // [8192]

    // output layout (flat, return order)
    float* out_q    = (float*)d_out;                       // 8388608
    float* out_enc  = out_q   + (size_t)BN * DIM;          // 134217728
    float* out_loss = out_enc + (size_t)BN * KCODES;       // 1
    float* out_ncs  = out_loss + 1;                        // 8192
    float* out_emaw = out_ncs  + KCODES;                   // 4194304
    float* out_w    = out_emaw + (size_t)KCODES * DIM;     // 4194304

    int*   ws_idx   = (int*)d_ws;                          // 16384
    float* ws_enorm = (float*)d_ws + BN;                   // 8192
    float* ws_sq    = (float*)d_ws + BN + KCODES;          // 1

    // 1. zero one-hot encodings (512 MB)
    zero_f4_kernel<<<4096, 256, 0, stream>>>((float4*)out_enc,
                                             (long long)BN * KCODES / 4);
    // 2. EMA decay pre-init (accumulators for the scatter phase)
    scale_kernel<<<32, 256, 0, stream>>>(ecs, out_ncs, KCODES);
    scale_kernel<<<16384, 256, 0, stream>>>(emaw, out_emaw, (long long)KCODES * DIM);
    // 3. codebook norms (+ zero loss accumulator)
    enorm_kernel<<<(KCODES * 32) / 256, 256, 0, stream>>>(emb, ws_enorm, ws_sq);
    // 4. fused fp32-WMMA distance GEMM + argmin
    const int lds_bytes = M_PER_WG * LDS_STRIDE * 4;       // 264192 B of 320 KB/WGP
    hipFuncSetAttribute((const void*)vq_argmin_kernel,
                        hipFuncAttributeMaxDynamicSharedMemorySize, lds_bytes);
    vq_argmin_kernel<<<BN / M_PER_WG, 256, lds_bytes, stream>>>(flat, emb, ws_enorm, ws_idx);
    // 5. gather quantized, loss partials, scatter one-hot / counts / dw
    quantize_scatter_kernel<<<BN, 256, 0, stream>>>(flat, emb, ws_idx, out_q, out_enc,
                                                    out_ncs, out_emaw, ws_sq);
    // 6. Laplace-normalize cluster sizes + finalize loss
    finalize_kernel<<<1, 1024, 0, stream>>>(out_ncs, ws_sq, out_loss);
    // 7. new_weight = new_ema_w / new_cluster_size
    new_weight_kernel<<<(KCODES * DIM) / 256, 256, 0, stream>>>(out_emaw, out_ncs, out_w);
}